// DecoderAttention_79542794322636
// MI455X (gfx1250) — compile-verified
//
#include <hip/hip_runtime.h>
#include <hip/hip_bf16.h>

typedef __attribute__((ext_vector_type(2))) float v2f;
typedef __attribute__((ext_vector_type(4))) float v4f;
typedef __attribute__((ext_vector_type(8))) float v8f;

#define B_  32
#define T_  4096
#define D_  1024
#define NC_ 16            // chunks per batch
#define TC_ (T_ / NC_)    // 256 rows per block
#define EPS_ 1e-8f

// ---------------------------------------------------------------------------
// Kernel 1: qW = h_d_t @ W  (32x1024x1024 f32 GEMM) via V_WMMA_F32_16X16X4_F32
// 128 output tiles of 16x16 (mtile in {0,1}, ntile in 0..63); one wave per tile.
// grid = 32 blocks x 128 threads (4 waves). EXEC all-ones everywhere (no
// divergence) as WMMA requires.
// ---------------------------------------------------------------------------
__global__ void qw_gemm_wmma(const float* __restrict__ A,   // h_d_t [32,1024]
                             const float* __restrict__ W,   // [1024,1024]
                             float* __restrict__ qW) {      // [32,1024]
  const int lane  = threadIdx.x & 31;
  const int wave  = threadIdx.x >> 5;
  const int tile  = blockIdx.x * 4 + wave;   // 0..127
  const int mtile = tile >> 6;               // 0..1
  const int ntile = tile & 63;               // 0..63
  const int half  = lane >> 4;               // 0: lanes 0-15, 1: lanes 16-31
  const int l15   = lane & 15;

  const int m = mtile * 16 + l15;            // A row (lanes 0-15 / 16-31 same M)
  const int n = ntile * 16 + l15;            // B/C column

  v8f c = {};
  for (int k0 = 0; k0 < D_; k0 += 4) {
    // A 16x4 f32 frag: lanes 0-15 hold K=k0,k0+1; lanes 16-31 hold K=k0+2,k0+3
    const int ka = k0 + half * 2;
    v2f a;
    a.x = A[m * D_ + ka];
    a.y = A[m * D_ + ka + 1];
    // B 4x16 f32 frag: VGPR0 -> rows K=k0 (lanes 0-15) / K=k0+2 (lanes 16-31),
    //                  VGPR1 -> rows K=k0+1 / K=k0+3
    v2f b;
    b.x = W[(size_t)ka * D_ + n];
    b.y = W[(size_t)(ka + 1) * D_ + n];
    c = __builtin_amdgcn_wmma_f32_16x16x4_f32(false, a, false, b,
                                              (short)0, c, false, false);
  }
  // C/D layout: VGPR r -> M=r (lanes 0-15), M=r+8 (lanes 16-31), N=lane%16
  #pragma unroll
  for (int r = 0; r < 8; ++r) {
    const int M = mtile * 16 + half * 8 + r;
    qW[(size_t)M * D_ + n] = c[r];
  }
}

// ---------------------------------------------------------------------------
// Kernel 2: fused single pass over h_d_all (the 512 MB stream; HBM-bound).
// For chunk of 256 rows: e[t] = qW[b].h[b,t];  partS += e;  partC += exp(e)*h.
// grid = (NC_, B_), 256 threads = 8 wave32s; each wave owns 32 rows.
// Each lane covers 32 of the 1024 d-positions (8 x b128, coalesced 512B/wave).
// h_d_all loads are NONTEMPORAL: 512 MB streamed once > 192 MB L2, so NT
// avoids thrashing L2 (keeps qW / partials resident) at full HBM rate.
// ---------------------------------------------------------------------------
__global__ void fused_pass(const float* __restrict__ h_all,  // [32,4096,1024]
                           const float* __restrict__ qW,     // [32,1024]
                           float* __restrict__ partC,        // [B_*NC_, 1024]
                           float* __restrict__ partS) {      // [B_*NC_]
  __shared__ float sacc[8 * D_];   // 32 KB
  __shared__ float sS[8];

  const int b     = blockIdx.y;
  const int chunk = blockIdx.x;
  const int lane  = threadIdx.x & 31;
  const int wave  = threadIdx.x >> 5;

  // qW slice for this lane's 32 d-positions (register resident)
  v4f q[8];
  #pragma unroll
  for (int i = 0; i < 8; ++i)
    q[i] = *(const v4f*)(qW + (size_t)b * D_ + i * 128 + lane * 4);

  v4f acc[8];
  #pragma unroll
  for (int i = 0; i < 8; ++i) acc[i] = (v4f){0.f, 0.f, 0.f, 0.f};
  float S = 0.f;

  const int rowsPerWave = TC_ / 8;                 // 32
  const int t0 = chunk * TC_ + wave * rowsPerWave;
  const float* base = h_all + (size_t)b * T_ * D_;

  for (int r = 0; r < rowsPerWave; ++r) {
    const float* row = base + (size_t)(t0 + r) * D_;
    v4f v[8];
    float dot = 0.f;
    #pragma unroll
    for (int i = 0; i < 8; ++i) {
      v[i] = __builtin_nontemporal_load(
                 (const v4f*)(row + i * 128 + lane * 4));
      dot += v[i].x * q[i].x + v[i].y * q[i].y +
             v[i].z * q[i].z + v[i].w * q[i].w;
    }
    // wave32 butterfly: all lanes end with full dot over D
    #pragma unroll
    for (int off = 16; off >= 1; off >>= 1)
      dot += __shfl_xor(dot, off, 32);
    S += dot;                       // raw-e denominator (lane-uniform)
    const float w = __expf(dot);
    #pragma unroll
    for (int i = 0; i < 8; ++i)
      acc[i] += w * v[i];
  }

  // per-wave partials -> LDS
  #pragma unroll
  for (int i = 0; i < 8; ++i)
    *(v4f*)(sacc + wave * D_ + i * 128 + lane * 4) = acc[i];
  if (lane == 0) sS[wave] = S;
  __syncthreads();

  // cross-wave reduce: thread handles 4 contiguous d-positions
  const int p = threadIdx.x * 4;
  v4f sum = (v4f){0.f, 0.f, 0.f, 0.f};
  #pragma unroll
  for (int w2 = 0; w2 < 8; ++w2)
    sum += *(const v4f*)(sacc + w2 * D_ + p);
  // partials are small (2 MB) and re-read next kernel: keep default RT policy
  *(v4f*)(partC + (size_t)(b * NC_ + chunk) * D_ + p) = sum;

  if (threadIdx.x == 0) {
    float s = 0.f;
    #pragma unroll
    for (int w2 = 0; w2 < 8; ++w2) s += sS[w2];
    partS[b * NC_ + chunk] = s;
  }
}

// ---------------------------------------------------------------------------
// Kernel 3: reduce NC_ partials per batch, normalize by 1/(S + eps).
// grid = B_ blocks x 256 threads; thread owns 4 d-positions.
// ---------------------------------------------------------------------------
__global__ void reduce_norm(const float* __restrict__ partC,
                            const float* __restrict__ partS,
                            float* __restrict__ out) {       // [32,1024]
  const int b = blockIdx.x;
  float S = 0.f;
  #pragma unroll
  for (int c = 0; c < NC_; ++c) S += partS[b * NC_ + c];
  const float scale = 1.0f / (S + EPS_);

  const int p = threadIdx.x * 4;
  v4f sum = (v4f){0.f, 0.f, 0.f, 0.f};
  #pragma unroll
  for (int c = 0; c < NC_; ++c)
    sum += *(const v4f*)(partC + (size_t)(b * NC_ + c) * D_ + p);
  *(v4f*)(out + (size_t)b * D_ + p) = sum * scale;
}

// ---------------------------------------------------------------------------
extern "C" void kernel_launch(void* const* d_in, const int* in_sizes, int n_in,
                              void* d_out, int out_size, void* d_ws, size_t ws_size,
                              hipStream_t stream) {
  const float* h_d_t   = (const float*)d_in[0];  // [32,1024]
  const float* h_d_all = (const float*)d_in[1];  // [32,4096,1024]
  const float* W       = (const float*)d_in[2];  // [1024,1024]
  float* out = (float*)d_out;                    // [32,1024]

  // workspace layout (floats): qW | partC | partS  (~2.2 MB total)
  float* qW    = (float*)d_ws;
  float* partC = qW + (size_t)B_ * D_;
  float* partS = partC + (size_t)B_ * NC_ * D_;

  qw_gemm_wmma<<<dim3(32), dim3(128), 0, stream>>>(h_d_t, W, qW);
  fused_pass<<<dim3(NC_, B_), dim3(256), 0, stream>>>(h_d_all, qW, partC, partS);
  reduce_norm<<<dim3(B_), dim3(256), 0, stream>>>(partC, partS, out);
}